// JacobiNet_5342939316736
// MI455X (gfx1250) — compile-verified
//
#include <hip/hip_runtime.h>

typedef float v2f __attribute__((ext_vector_type(2)));
typedef float v8f __attribute__((ext_vector_type(8)));

#define N_NODES 100000
#define N_EDGES 1600000
#define IN_CH   256
#define HID     64
#define OUT_CH  32
#define KORD    10
#define NLAYERS 2
#define NH      (N_NODES * HID)   // 6,400,000

// ---------------- degree / norm build ----------------

__global__ void k_init_deg(float* __restrict__ deg) {
  int i = blockIdx.x * blockDim.x + threadIdx.x;
  if (i < N_NODES) deg[i] = 1.0f;  // self loop contributes 1
}

__global__ void k_deg_edges(const int* __restrict__ ei, float* __restrict__ deg) {
  int e = blockIdx.x * blockDim.x + threadIdx.x;
  if (e < N_EDGES) atomicAdd(&deg[ei[N_EDGES + e]], 1.0f);  // col index
}

__global__ void k_dinv(float* __restrict__ deg) {
  int i = blockIdx.x * blockDim.x + threadIdx.x;
  if (i < N_NODES) {
    float d = deg[i];
    deg[i] = (d > 0.0f) ? rsqrtf(d) : 0.0f;
  }
}

__global__ void k_norm(const int* __restrict__ ei, const float* __restrict__ dinv,
                       float* __restrict__ norm) {
  int e = blockIdx.x * blockDim.x + threadIdx.x;
  if (e < N_EDGES) norm[e] = dinv[ei[e]] * dinv[ei[N_EDGES + e]];
}

// ---------------- WMMA GEMMs (fp32, 16x16x4) ----------------
// One wave computes one 16x16 tile. A: lanes0-15 K={0,1}, lanes16-31 K={2,3}.
// B: VGPR r holds row (r + 2*half). C/D: VGPR r -> M = r + 8*half, N = lane%16.

__global__ __launch_bounds__(128) void k_gemm1(const float* __restrict__ x,
                                               const float* __restrict__ W,
                                               const float* __restrict__ bias,
                                               float* __restrict__ h) {
  int wave = threadIdx.x >> 5;          // 4 waves -> 4 col tiles (HID=64)
  int lane = threadIdx.x & 31;
  int half = lane >> 4;
  int lr   = lane & 15;
  int m0 = blockIdx.x << 4;
  int n0 = wave << 4;
  v8f acc = {};
  const float* arow = x + (size_t)(m0 + lr) * IN_CH;
  for (int kk = 0; kk < IN_CH; kk += 4) {
    v2f a = *(const v2f*)(arow + kk + 2 * half);
    v2f b;
    b.x = W[(size_t)(kk + 2 * half)     * HID + n0 + lr];
    b.y = W[(size_t)(kk + 2 * half + 1) * HID + n0 + lr];
    acc = __builtin_amdgcn_wmma_f32_16x16x4_f32(false, a, false, b,
                                                (short)0, acc, false, false);
  }
  float bv = bias[n0 + lr];
#pragma unroll
  for (int r = 0; r < 8; ++r) {
    int row = m0 + r + 8 * half;
    float v = acc[r] + bv;
    h[(size_t)row * HID + n0 + lr] = fmaxf(v, 0.0f);
  }
}

// final GEMM with relu fused into the A-fragment load: out = relu(pre) @ W2 + b2
__global__ __launch_bounds__(64) void k_gemm2(const float* __restrict__ pre,
                                              const float* __restrict__ W,
                                              const float* __restrict__ bias,
                                              float* __restrict__ out) {
  int wave = threadIdx.x >> 5;          // 2 waves -> 2 col tiles (OUT_CH=32)
  int lane = threadIdx.x & 31;
  int half = lane >> 4;
  int lr   = lane & 15;
  int m0 = blockIdx.x << 4;
  int n0 = wave << 4;
  v8f acc = {};
  const float* arow = pre + (size_t)(m0 + lr) * HID;
  for (int kk = 0; kk < HID; kk += 4) {
    v2f a = *(const v2f*)(arow + kk + 2 * half);
    a.x = fmaxf(a.x, 0.0f);             // h = relu(pre), fused
    a.y = fmaxf(a.y, 0.0f);
    v2f b;
    b.x = W[(size_t)(kk + 2 * half)     * OUT_CH + n0 + lr];
    b.y = W[(size_t)(kk + 2 * half + 1) * OUT_CH + n0 + lr];
    acc = __builtin_amdgcn_wmma_f32_16x16x4_f32(false, a, false, b,
                                                (short)0, acc, false, false);
  }
  float bv = bias[n0 + lr];
#pragma unroll
  for (int r = 0; r < 8; ++r) {
    int row = m0 + r + 8 * half;
    out[(size_t)row * OUT_CH + n0 + lr] = acc[r] + bv;  // no relu on final
  }
}

// ---------------- SpMM propagate ----------------

// dst[col[e]] += norm[e] * src[row[e]]   -- one wave per edge, 2 channels/lane
__global__ __launch_bounds__(256) void k_spmm_edges(const int* __restrict__ ei,
                                                    const float* __restrict__ norm,
                                                    const float* __restrict__ src,
                                                    float* __restrict__ dst) {
  int e = (blockIdx.x << 3) + (threadIdx.x >> 5);
  if (e >= N_EDGES) return;
  int lane = threadIdx.x & 31;
  int r = ei[e];
  int c = ei[N_EDGES + e];
  float nv = norm[e];
  const float* s = src + (size_t)r * HID;
  float*       d = dst + (size_t)c * HID;
  atomicAdd(d + lane,      nv * s[lane]);
  atomicAdd(d + lane + 32, nv * s[lane + 32]);
}

// ---------------- fused elementwise kernels ----------------

// Layer start: v = src; optional relu; optional h materialization;
// outb = c0*v (may be in-place on src); f1dst = dinv^2 * v (self-loop init for
// the first propagate of this layer). One read pass instead of three.
__global__ void k_layer_start(const float* src, float* outb, float* hdst,
                              float* f1dst, const float* __restrict__ dinv,
                              const float* __restrict__ coeffs, int cidx,
                              int do_relu) {
  int idx = blockIdx.x * blockDim.x + threadIdx.x;
  if (idx < NH) {
    float v = src[idx];
    if (do_relu) v = fmaxf(v, 0.0f);
    if (hdst) hdst[idx] = v;
    float di = dinv[idx >> 6];
    f1dst[idx] = di * di * v;
    outb[idx] = coeffs[cidx] * v;
  }
}

// Fused Jacobi combine + next-propagate self-loop init:
//   p = An*P + Bn*T1 + Cn*T0 ; P = p ; outb += c[cidx]*p ;
//   if (nxt) nxt = dinv^2 * p   (nxt may alias T0: read-then-write per element)
__global__ void k_combine_init(float* P, const float* T1, const float* T0,
                               float* outb, float An, float Bn, float Cn,
                               const float* __restrict__ coeffs, int cidx,
                               float* nxt, const float* __restrict__ dinv) {
  int idx = blockIdx.x * blockDim.x + threadIdx.x;
  if (idx < NH) {
    float p = An * P[idx] + Bn * T1[idx] + Cn * T0[idx];
    P[idx] = p;
    outb[idx] += coeffs[cidx] * p;
    if (nxt) {
      float di = dinv[idx >> 6];
      nxt[idx] = di * di * p;
    }
  }
}

// ---------------- host orchestration ----------------

extern "C" void kernel_launch(void* const* d_in, const int* in_sizes, int n_in,
                              void* d_out, int out_size, void* d_ws, size_t ws_size,
                              hipStream_t stream) {
  const float* x      = (const float*)d_in[0];
  const int*   ei     = (const int*)  d_in[1];
  const float* W1     = (const float*)d_in[2];
  const float* b1     = (const float*)d_in[3];
  const float* coeffs = (const float*)d_in[4];
  const float* W2     = (const float*)d_in[5];
  const float* b2     = (const float*)d_in[6];
  float* out = (float*)d_out;

  char* ws = (char*)d_ws;
  size_t off = 0;
  auto alloc = [&](size_t bytes) -> float* {
    float* p = (float*)(ws + off);
    off += (bytes + 255) & ~(size_t)255;
    return p;
  };
  float* dinv = alloc((size_t)N_NODES * 4);
  float* norm = alloc((size_t)N_EDGES * 4);
  float* X0   = alloc((size_t)NH * 4);
  float* X1   = alloc((size_t)NH * 4);
  float* X2   = alloc((size_t)NH * 4);
  float* OUTB = alloc((size_t)NH * 4);

  const int gN  = (N_NODES + 255) / 256;
  const int gE  = (N_EDGES + 255) / 256;
  const int gNH = (NH + 255) / 256;
  const int gEdgeWave = (N_EDGES + 7) / 8;   // 8 edges (waves) per 256-thread block
  const int gTile = N_NODES / 16;            // 6250 row tiles

  // norm build (deg stored in dinv buffer, transformed in place)
  k_init_deg <<<gN, 256, 0, stream>>>(dinv);
  k_deg_edges<<<gE, 256, 0, stream>>>(ei, dinv);
  k_dinv     <<<gN, 256, 0, stream>>>(dinv);
  k_norm     <<<gE, 256, 0, stream>>>(ei, dinv, norm);

  // h = relu(x @ W1 + b1)  -> X0
  k_gemm1<<<gTile, 128, 0, stream>>>(x, W1, b1, X0);

  float* h = X0; float* f1 = X1; float* f2 = X2;
  const double aj = 0.5, bj = 0.5;

  for (int l = 0; l < NLAYERS; ++l) {
    int cbase = l * (KORD + 1);
    // out = c0*h ; f1 = dinv^2*h (self-loop init). For l>0 this also applies
    // relu to OUTB (in place) and materializes h into the free buffer.
    if (l == 0) {
      k_layer_start<<<gNH, 256, 0, stream>>>(h, OUTB, (float*)nullptr, f1,
                                             dinv, coeffs, cbase + 0, 0);
    } else {
      k_layer_start<<<gNH, 256, 0, stream>>>(OUTB, OUTB, h, f1,
                                             dinv, coeffs, cbase + 0, 1);
    }
    // Tx1 = propagate(h) -> f1 ; out += c1*Tx1 ; f2 = dinv^2*Tx1 (init for k=2)
    k_spmm_edges<<<gEdgeWave, 256, 0, stream>>>(ei, norm, h, f1);
    k_combine_init<<<gNH, 256, 0, stream>>>(f1, f1, f1, OUTB,
                                            1.0f, 0.0f, 0.0f,
                                            coeffs, cbase + 1, f2, dinv);
    float* T0 = h; float* T1 = f1; float* FR = f2;
    for (int k = 2; k <= KORD; ++k) {
      double n = (double)(k - 1);
      double An = (2*n + aj + bj + 1) * (2*n + aj + bj + 2) /
                  (2 * (n + 1) * (n + aj + bj + 1));
      double Bn = (aj*aj - bj*bj) * (2*n + aj + bj + 1) /
                  (2 * (n + 1) * (n + aj + bj + 1) * (2*n + aj + bj));
      double Cn = (n + aj) * (n + bj) * (2*n + aj + bj + 2) /
                  ((n + 1) * (n + aj + bj + 1) * (2*n + aj + bj));
      // P = propagate(T1) -> FR  (FR was pre-initialized with self-loop term)
      k_spmm_edges<<<gEdgeWave, 256, 0, stream>>>(ei, norm, T1, FR);
      // Tx2 = An*P + Bn*T1 - Cn*T0 (in place in FR); out += c_k*Tx2;
      // pre-init the NEXT propagate destination (buffer T0, which rotates to FR)
      float* nxt = (k < KORD) ? T0 : (float*)nullptr;
      k_combine_init<<<gNH, 256, 0, stream>>>(FR, T1, T0, OUTB,
                                              (float)An, (float)Bn, (float)(-Cn),
                                              coeffs, cbase + k, nxt, dinv);
      float* tmp = T0; T0 = T1; T1 = FR; FR = tmp;  // rotate
    }
    // hand free buffers to the next layer (h materialized by k_layer_start)
    h = FR; f1 = T0; f2 = T1;
  }

  // final: d_out = relu(OUTB) @ W2 + b2  (relu fused into GEMM2 A-loads)
  k_gemm2<<<gTile, 64, 0, stream>>>(OUTB, W2, b2, out);

  (void)in_sizes; (void)n_in; (void)out_size; (void)ws_size;
}